// VoxelEncoder_90022514524738
// MI455X (gfx1250) — compile-verified
//
#include <hip/hip_runtime.h>
#include <math.h>

// Problem constants (fixed by the reference)
#define GXD 30
#define GYD 40
#define GZD 40
#define GRID_ (GXD*GYD*GZD)   // 48000
#define MAXV 2000
#define MAXP 10
#define NPTS 100000
#define NB   16
#define NF   101

typedef __attribute__((ext_vector_type(2))) float v2f;
typedef __attribute__((ext_vector_type(8))) float v8f;

// ---------------------------------------------------------------- init scratch
__global__ void vox_initK(int* __restrict__ first_pt, int* __restrict__ cnt,
                          int* __restrict__ thr, int* __restrict__ rankv,
                          int* __restrict__ cand, int total) {
    int i = blockIdx.x * blockDim.x + threadIdx.x;
    if (i < total) {
        first_pt[i] = NPTS;
        cnt[i]      = 0;
        thr[i]      = -1;
        rankv[i]    = 0x7fffffff;   // "unranked"
        cand[i]     = NPTS;         // empty candidate
    }
}

// ------------------------------------------------- pass 1: voxelize, count, min
__global__ void vox_pointK(const float* __restrict__ pc, int* __restrict__ vid_pt,
                           int* __restrict__ first_pt, int* __restrict__ cnt) {
    int p = blockIdx.x * blockDim.x + threadIdx.x;
    int b = blockIdx.y;
    if (p >= NPTS) return;
    const float* q = pc + ((size_t)b * NPTS + p) * NF;
    float x = q[0], y = q[1], z = q[2];
    int cx = (int)floorf((x + 1.0f) / 0.1f);
    int cy = (int)floorf((y + 1.0f) / 0.1f);
    int cz = (int)floorf((z + 1.0f) / 0.1f);
    bool ok = (cx >= 0) & (cx < GXD) & (cy >= 0) & (cy < GYD) & (cz >= 0) & (cz < GZD);
    int vid = ok ? (cx * GYD + cy) * GZD + cz : GRID_;
    vid_pt[(size_t)b * NPTS + p] = vid;
    if (ok) {
        atomicMin(&first_pt[b * GRID_ + vid], p);
        atomicAdd(&cnt[b * GRID_ + vid], 1);
    }
}

// ------------------- pass 2: rank voxels by first-appearance (prefix sum scan)
// 8 elems/thread serial + wave32 shfl scan + 32-entry LDS scan: ~3 barriers/chunk.
#define SCAN_E 8
__global__ void __launch_bounds__(1024)
vox_scanK(const int* __restrict__ vid_pt, const int* __restrict__ first_pt,
          int* __restrict__ rankv, int* __restrict__ vox, int* __restrict__ nvox) {
    int b = blockIdx.x, t = threadIdx.x;
    int lane = t & 31, wid = t >> 5;
    __shared__ int swave[32];
    __shared__ int carry;
    if (t == 0) carry = 0;
    __syncthreads();
    const int CH = 1024 * SCAN_E;
    for (int base = 0; base < NPTS; base += CH) {
        int p0 = base + t * SCAN_E;
        int vv[SCAN_E], flg[SCAN_E], loc[SCAN_E];
        int s = 0;
        #pragma unroll
        for (int i = 0; i < SCAN_E; i++) {
            int p = p0 + i;
            int isf = 0, v = GRID_;
            if (p < NPTS) {
                v = vid_pt[(size_t)b * NPTS + p];
                if (v < GRID_ && first_pt[b * GRID_ + v] == p) isf = 1;
            }
            vv[i] = v; flg[i] = isf; loc[i] = s; s += isf;
        }
        // wave32 inclusive scan of per-thread sums
        int inc = s;
        #pragma unroll
        for (int off = 1; off < 32; off <<= 1) {
            int o = __shfl_up(inc, off, 32);
            if (lane >= off) inc += o;
        }
        if (lane == 31) swave[wid] = inc;
        __syncthreads();
        if (wid == 0) {
            int x = swave[lane];
            #pragma unroll
            for (int off = 1; off < 32; off <<= 1) {
                int o = __shfl_up(x, off, 32);
                if (lane >= off) x += o;
            }
            swave[lane] = x;   // inclusive scan of wave sums
        }
        __syncthreads();
        int waveoff = (wid > 0) ? swave[wid - 1] : 0;
        int exclT = carry + waveoff + (inc - s);   // exclusive prefix of this thread
        #pragma unroll
        for (int i = 0; i < SCAN_E; i++) {
            if (flg[i]) {
                int r = exclT + loc[i];
                rankv[b * GRID_ + vv[i]] = r;
                if (r < MAXV) vox[b * MAXV + r] = vv[i];
            }
        }
        __syncthreads();
        if (t == 0) carry += swave[31];
        __syncthreads();
    }
    if (t == 0) { int tt = carry; nvox[b] = (tt < MAXV) ? tt : MAXV; }
}

// ------- fold "ranked" predicate into vid_pt so the 10 passes skip rankv reads
__global__ void vox_flagK(int* __restrict__ vid_pt, const int* __restrict__ rankv) {
    int p = blockIdx.x * blockDim.x + threadIdx.x;
    int b = blockIdx.y;
    if (p >= NPTS) return;
    size_t ip = (size_t)b * NPTS + p;
    int v = vid_pt[ip];
    if (v < GRID_ && rankv[b * GRID_ + v] >= MAXV) vid_pt[ip] = GRID_;
}

// ------------- pass 3 (x10): k-th smallest point index per ranked voxel
__global__ void vox_selK(const int* __restrict__ vid_pt, const int* __restrict__ thr,
                         int* __restrict__ cand) {
    int p = blockIdx.x * blockDim.x + threadIdx.x;
    int b = blockIdx.y;
    if (p >= NPTS) return;
    int v = vid_pt[(size_t)b * NPTS + p];
    if (v < GRID_) {
        int idx = b * GRID_ + v;
        if (p > thr[idx]) atomicMin(&cand[idx], p);
    }
}

__global__ void vox_commitK(const int* __restrict__ nvox, const int* __restrict__ vox,
                            int* __restrict__ thr, int* __restrict__ cand,
                            int* __restrict__ sel, int k) {
    int r = blockIdx.x * blockDim.x + threadIdx.x;
    int b = blockIdx.y;
    if (r >= MAXV) return;
    if (r >= nvox[b]) return;
    int v = vox[b * MAXV + r];
    int idx = b * GRID_ + v;
    int c = cand[idx];
    if (c < NPTS) {
        sel[((size_t)b * MAXV + r) * MAXP + k] = c;
        thr[idx] = c;
        cand[idx] = NPTS;   // reset for next round
    }
}

// ---- pass 4: gather + WMMA reduction (D = onesA(16x4) x featB(4x16) + C)
// One wave32 per (rank, batch). Branchless: row pointers hoisted & clamped,
// loads unconditional, masked to zero with selects. All-ones A keeps exact f32
// accumulation and makes the hardware K-slot mapping irrelevant.
__global__ void __launch_bounds__(32)
vox_gatherK(const float* __restrict__ pc, const int* __restrict__ nvox,
            const int* __restrict__ vox, const int* __restrict__ cnt,
            const int* __restrict__ sel, float* __restrict__ enc,
            int* __restrict__ coords, int* __restrict__ npts_out) {
    int r = blockIdx.x;
    int b = blockIdx.y;
    int lane = threadIdx.x;
    float* e  = enc + ((size_t)b * MAXV + r) * NF;
    int*   co = coords + ((size_t)b * MAXV + r) * 3;
    int*   np = npts_out + b * MAXV + r;

    if (r >= nvox[b]) {                      // empty rank: write zeros everywhere
        for (int f = lane; f < NF; f += 32) e[f] = 0.0f;
        if (lane == 0) { co[0] = 0; co[1] = 0; co[2] = 0; *np = 0; }
        return;
    }

    int v = vox[b * MAXV + r];
    int n = cnt[b * GRID_ + v];
    if (n > MAXP) n = MAXP;
    const int*   sp  = sel + ((size_t)b * MAXV + r) * MAXP;
    const float* pcb = pc + (size_t)b * NPTS * NF;
    float scale = 10.0f / ((float)n * (float)n);   // encoded = 10*sum/n^2

    int nb_ = lane & 15;
    int hi2 = (lane >> 4) << 1;   // 0 or 2

    // Per-lane row pointers for its 6 K slots, clamped to valid sel entries.
    const float* rowA[3]; const float* rowB[3];
    bool mA[3], mB[3];
    #pragma unroll
    for (int kc = 0; kc < 3; kc++) {
        int kA = kc * 4 + hi2, kB = kA + 1;
        int iA = sp[(kA < n) ? kA : (n - 1)];
        int iB = sp[(kB < n) ? kB : (n - 1)];
        rowA[kc] = pcb + (size_t)iA * NF;
        rowB[kc] = pcb + (size_t)iB * NF;
        mA[kc] = (kA < n);
        mB[kc] = (kB < n);
    }

    v2f a; a.x = 1.0f; a.y = 1.0f;                 // all-ones A matrix
    for (int c = 0; c < 7; c++) {                  // 7 chunks of 16 features
        int f = c * 16 + nb_;
        bool fok = (f < NF);
        int fc = fok ? f : 0;                      // clamp (avoid OOB on tail chunk)
        v8f acc = {0.f, 0.f, 0.f, 0.f, 0.f, 0.f, 0.f, 0.f};
        #pragma unroll
        for (int kc = 0; kc < 3; kc++) {
            float bx = rowA[kc][fc];               // unconditional loads
            float by = rowB[kc][fc];
            bx = (mA[kc] & fok) ? bx : 0.0f;       // branchless zeroing
            by = (mB[kc] & fok) ? by : 0.0f;
            v2f bb; bb.x = bx; bb.y = by;
            acc = __builtin_amdgcn_wmma_f32_16x16x4_f32(
                false, a, false, bb, (short)0, acc, false, false);
        }
        if ((lane < 16) & fok) e[f] = acc[0] * scale;  // M=0 row holds the sum
    }
    if (lane == 0) {
        *np = n;
        co[0] = v % GZD;                // z
        co[1] = (v / GZD) % GYD;        // y
        co[2] = v / (GZD * GYD);        // x
    }
}

// ----------------------------------------------------------------------------
extern "C" void kernel_launch(void* const* d_in, const int* in_sizes, int n_in,
                              void* d_out, int out_size, void* d_ws, size_t ws_size,
                              hipStream_t stream) {
    (void)in_sizes; (void)n_in; (void)out_size; (void)ws_size;
    const float* pc = (const float*)d_in[0];

    // Workspace carve-up (~23.2 MB, all int32)
    char* w = (char*)d_ws;
    size_t gcount = (size_t)NB * GRID_;
    int* first_pt = (int*)w;  w += gcount * 4;
    int* cnt      = (int*)w;  w += gcount * 4;
    int* thr      = (int*)w;  w += gcount * 4;
    int* rankv    = (int*)w;  w += gcount * 4;
    int* cand     = (int*)w;  w += gcount * 4;
    int* vid_pt   = (int*)w;  w += (size_t)NB * NPTS * 4;
    int* vox      = (int*)w;  w += (size_t)NB * MAXV * 4;
    int* sel      = (int*)w;  w += (size_t)NB * MAXV * MAXP * 4;
    int* nvox     = (int*)w;  w += (size_t)NB * 4;

    // Output layout: enc (B*MAXV*NF f32) | coords (B*MAXV*3 i32) | npts (B*MAXV i32)
    float* enc    = (float*)d_out;
    int*   coords = (int*)(enc + (size_t)NB * MAXV * NF);
    int*   nptr   = coords + (size_t)NB * MAXV * 3;

    const int tb = 256;
    vox_initK<<<dim3((unsigned)((gcount + tb - 1) / tb)), tb, 0, stream>>>(
        first_pt, cnt, thr, rankv, cand, (int)gcount);
    vox_pointK<<<dim3((NPTS + tb - 1) / tb, NB), tb, 0, stream>>>(
        pc, vid_pt, first_pt, cnt);
    vox_scanK<<<dim3(NB), 1024, 0, stream>>>(vid_pt, first_pt, rankv, vox, nvox);
    vox_flagK<<<dim3((NPTS + tb - 1) / tb, NB), tb, 0, stream>>>(vid_pt, rankv);
    for (int k = 0; k < MAXP; k++) {
        vox_selK<<<dim3((NPTS + tb - 1) / tb, NB), tb, 0, stream>>>(
            vid_pt, thr, cand);
        vox_commitK<<<dim3((MAXV + tb - 1) / tb, NB), tb, 0, stream>>>(
            nvox, vox, thr, cand, sel, k);
    }
    vox_gatherK<<<dim3(MAXV, NB), 32, 0, stream>>>(
        pc, nvox, vox, cnt, sel, enc, coords, nptr);
}